// PairwiseRelationModuleLight_81415400063717
// MI455X (gfx1250) — compile-verified
//
#include <hip/hip_runtime.h>
#include <hip/hip_bf16.h>
#include <cstddef>

// Problem dimensions (fixed by the reference)
static constexpr int kB = 4;
static constexpr int kN = 256;
static constexpr int kD = 256;
static constexpr int kL = 256;
static constexpr int kH = 128;
// PAIR_DIM = 2*kD + 3 + kL = 771 (W1 is row-major [PAIR_DIM, kH])

typedef __attribute__((ext_vector_type(2))) float v2f;
typedef __attribute__((ext_vector_type(8))) float v8f;

// ---------------------------------------------------------------------------
// Kernel 1: s_lang[b,h] = sum_l lang[b,l] * W1_lang[l,h]   (tiny: 512 dots)
// grid = kB, block = kH
// ---------------------------------------------------------------------------
__global__ void k_slang(const float* __restrict__ lang,
                        const float* __restrict__ W1,
                        float* __restrict__ slang) {
    const int b = blockIdx.x;
    const int h = threadIdx.x;
    const float* Wl = W1 + (size_t)(2 * kD + 3) * kH; // rows [2D+3, 2D+3+L)
    float acc = 0.0f;
#pragma unroll 4
    for (int l = 0; l < kL; ++l) {
        acc = fmaf(lang[b * kL + l], Wl[(size_t)l * kH + h], acc);
    }
    slang[b * kH + h] = acc;
}

// ---------------------------------------------------------------------------
// Kernel 2 (WMMA): dual GEMM sharing the A fragment.
//   U0 = X @ W1_i, V0 = X @ W1_j  with X = object_features as (B*N, D).
//   One wave owns one 16x16 tile of BOTH U and V (two accumulators, two
//   v_wmma_f32_16x16x4_f32 per K-step, independent -> no RAW hazard).
//   Epilogue folds in t = (centers/5)@W1_rel, s_lang, b1:
//     u = U0 + t + s_lang + b1 ;  v = V0 - t
//   Tiles: (1024/16) x (128/16) = 64 x 8 = 512. 8 waves/block -> 64 blocks.
// ---------------------------------------------------------------------------
__global__ void k_uv_wmma(const float* __restrict__ obj,
                          const float* __restrict__ centers,
                          const float* __restrict__ W1,
                          const float* __restrict__ b1,
                          const float* __restrict__ slang,
                          float* __restrict__ u,
                          float* __restrict__ v) {
    const int wave = threadIdx.x >> 5;
    const int lane = threadIdx.x & 31;
    const int tile = blockIdx.x * 8 + wave;   // 0..511
    const int tm   = tile >> 3;               // row tile (bn direction), 0..63
    const int tn   = tile & 7;                // col tile (h direction), 0..7
    const int half = lane >> 4;
    const int l15  = lane & 15;

    const int arow = tm * 16 + l15;           // A-matrix row (bn), both halves
    const int bcol = tn * 16 + l15;           // B-matrix col (h)

    v8f cu = {};
    v8f cv = {};
    for (int k = 0; k < kD; k += 4) {
        const int ka = k + 2 * half;          // this half's K pair {ka, ka+1}
        v2f a, bi, bj;
        a.x  = obj[(size_t)arow * kD + ka];
        a.y  = obj[(size_t)arow * kD + ka + 1];
        bi.x = W1[(size_t)ka * kH + bcol];              // W1_i rows [0, D)
        bi.y = W1[(size_t)(ka + 1) * kH + bcol];
        bj.x = W1[(size_t)(kD + ka) * kH + bcol];       // W1_j rows [D, 2D)
        bj.y = W1[(size_t)(kD + ka + 1) * kH + bcol];
        cu = __builtin_amdgcn_wmma_f32_16x16x4_f32(
                 false, a, false, bi, (short)0, cu, false, false);
        cv = __builtin_amdgcn_wmma_f32_16x16x4_f32(
                 false, a, false, bj, (short)0, cv, false, false);
    }

#pragma unroll
    for (int r = 0; r < 8; ++r) {
        const int row = tm * 16 + r + 8 * half;   // bn index
        const int col = tn * 16 + l15;            // h index
        const int b   = row >> 8;                 // row / kN
        float t = 0.0f;
#pragma unroll
        for (int c3 = 0; c3 < 3; ++c3) {
            const float rc = centers[(size_t)row * 3 + c3] / 5.0f;
            t = fmaf(rc, W1[(size_t)(2 * kD + c3) * kH + col], t);
        }
        u[(size_t)row * kH + col] = cu[r] + t + slang[b * kH + col] + b1[col];
        v[(size_t)row * kH + col] = cv[r] - t;
    }
}

// ---------------------------------------------------------------------------
// Kernel 3: scores + mask + softmax, fully fused (VALU by design: the relu
// breaks the h-contraction into elementwise work that WMMA can't absorb
// without materializing the 134MB hidden tensor).
// grid = kB*kN (one block per (b,n)), block = kN threads (thread = j)
// ---------------------------------------------------------------------------
__global__ void k_scores(const float* __restrict__ u,
                         const float* __restrict__ v,
                         const float* __restrict__ W2,
                         const float* __restrict__ b2,
                         const int* __restrict__ mask,
                         float* __restrict__ wout) {
    __shared__ float us[kH];
    __shared__ float w2s[kH];
    __shared__ float red[kN];

    const int bn = blockIdx.x;
    const int b  = bn / kN;
    const int n  = bn % kN;
    const int j  = threadIdx.x;

    if (j < kH) {
        us[j]  = u[(size_t)bn * kH + j];
        w2s[j] = W2[j];
    }
    __syncthreads();

    const float* vr = v + ((size_t)b * kN + j) * kH;
    float s = b2[0];
#pragma unroll 4
    for (int h = 0; h < kH; ++h) {
        const float hv = us[h] + vr[h];
        s = fmaf(fmaxf(hv, 0.0f), w2s[h], s);
    }

    const bool m = (mask[b * kN + n] != 0) && (mask[b * kN + j] != 0);
    s = m ? s : -1.0e9f;

    // block-wide max reduction over j
    red[j] = s;
    __syncthreads();
    for (int off = kN / 2; off > 0; off >>= 1) {
        if (j < off) red[j] = fmaxf(red[j], red[j + off]);
        __syncthreads();
    }
    const float mx = red[0];
    __syncthreads();

    const float e = __expf(s - mx);
    red[j] = e;
    __syncthreads();
    for (int off = kN / 2; off > 0; off >>= 1) {
        if (j < off) red[j] += red[j + off];
        __syncthreads();
    }
    wout[(size_t)bn * kN + j] = e / red[0];
}

// ---------------------------------------------------------------------------
// Kernel 4 (WMMA): enhanced[b] = obj[b] + weights[b] @ obj[b]  (fp32 GEMM)
//   One wave per 16x16 C tile, V_WMMA_F32_16X16X4_F32, K-loop 64 x (K=4).
// grid = 128 blocks x 256 threads (8 waves/block, 1024 tiles total)
// ---------------------------------------------------------------------------
__global__ void k_context(const float* __restrict__ wgt,
                          const float* __restrict__ obj,
                          float* __restrict__ out) {
    const int wave = threadIdx.x >> 5;
    const int lane = threadIdx.x & 31;
    const int tile = blockIdx.x * 8 + wave;   // 0..1023
    const int b    = tile >> 8;               // 256 tiles per batch
    const int t    = tile & 255;
    const int tm   = t >> 4;                  // tile row (n direction)
    const int tn   = t & 15;                  // tile col (d direction)
    const int half = lane >> 4;
    const int l15  = lane & 15;

    const float* Wb = wgt + (size_t)b * kN * kN;
    const float* Fb = obj + (size_t)b * kN * kD;

    v8f c = {};
    const int arow = tm * 16 + l15;           // A-matrix row M (both halves)
    const int bcol = tn * 16 + l15;           // B-matrix col N

    for (int k = 0; k < kN; k += 4) {
        const int ka = k + 2 * half;          // this half's K pair {ka, ka+1}
        v2f a, bb;
        a.x  = Wb[(size_t)arow * kN + ka];
        a.y  = Wb[(size_t)arow * kN + ka + 1];
        bb.x = Fb[(size_t)ka * kD + bcol];
        bb.y = Fb[(size_t)(ka + 1) * kD + bcol];
        c = __builtin_amdgcn_wmma_f32_16x16x4_f32(
                false, a, false, bb, (short)0, c, false, false);
    }

#pragma unroll
    for (int r = 0; r < 8; ++r) {
        const int row = tm * 16 + r + 8 * half;
        const int col = tn * 16 + l15;
        const size_t idx = (size_t)row * kD + col;
        out[(size_t)b * kN * kD + idx] = Fb[idx] + c[r];
    }
}

// ---------------------------------------------------------------------------
// Host-side launch
// Inputs (setup_inputs order):
//   0 object_features (B,N,D) f32   1 language_embedding (B,L) f32
//   2 centers (B,N,3) f32           3 object_mask (B,N) int
//   4 W1 (771,H) f32                5 b1 (H) f32
//   6 W2 (H,1) f32                  7 b2 (1) f32
// Output: enhanced_features (B,N,D) f32 ++ relation_weights (B,N,N) f32
// ---------------------------------------------------------------------------
extern "C" void kernel_launch(void* const* d_in, const int* in_sizes, int n_in,
                              void* d_out, int out_size, void* d_ws, size_t ws_size,
                              hipStream_t stream) {
    const float* obj     = (const float*)d_in[0];
    const float* lang    = (const float*)d_in[1];
    const float* centers = (const float*)d_in[2];
    const int*   mask    = (const int*)d_in[3];
    const float* W1      = (const float*)d_in[4];
    const float* b1      = (const float*)d_in[5];
    const float* W2      = (const float*)d_in[6];
    const float* b2      = (const float*)d_in[7];

    float* enhanced = (float*)d_out;                          // B*N*D floats
    float* weights  = (float*)d_out + (size_t)kB * kN * kD;   // B*N*N floats

    float* ws    = (float*)d_ws;
    float* u     = ws;                                        // B*N*H
    float* v     = u + (size_t)kB * kN * kH;                  // B*N*H
    float* slang = v + (size_t)kB * kN * kH;                  // B*H

    k_slang<<<kB, kH, 0, stream>>>(lang, W1, slang);
    k_uv_wmma<<<64, 256, 0, stream>>>(obj, centers, W1, b1, slang, u, v);
    k_scores<<<kB * kN, kN, 0, stream>>>(u, v, W2, b2, mask, weights);
    k_context<<<128, 256, 0, stream>>>(weights, obj, enhanced);
}